// GNNLayer_2121713845031
// MI455X (gfx1250) — compile-verified
//
#include <hip/hip_runtime.h>
#include <hip/hip_bf16.h>

typedef __attribute__((ext_vector_type(2))) float v2f;
typedef __attribute__((ext_vector_type(8))) float v8f;

#define FIN   128
#define HC_   128
#define HEADS 4
#define CH    32
#define NEG_SLOPE 0.2f

__device__ __forceinline__ float lrelu(float v) { return v > 0.f ? v : NEG_SLOPE * v; }

// ---------------------------------------------------------------------------
// Kernel 1: h[BN,128] = x[BN,128] @ W[128,128] via V_WMMA_F32_16X16X4_F32.
// W staged in LDS pre-swizzled as [k/2][col][k&1] so every B fragment is one
// aligned ds_load_b64 (kills the v_dual_mov packing chains).
// Block = 128 threads (4 waves); each wave owns a 16-row x 128-col strip.
// ---------------------------------------------------------------------------
__global__ __launch_bounds__(128) void k_gemm(const float* __restrict__ x,
                                              const float* __restrict__ W,
                                              float* __restrict__ h, int M) {
  __shared__ float sW[FIN * HC_];   // 64KB, swizzled: ((k>>1)*128 + col)*2 + (k&1)
  const int tid = threadIdx.x;
  for (int i = tid * 4; i < FIN * HC_; i += 128 * 4) {
    float4 wv = *(const float4*)(W + i);
    int k = i >> 7;          // row of W (K index)
    int c = i & 127;         // first col
    float* dp = sW + ((k >> 1) * 256 + (k & 1));
    dp[(c + 0) * 2] = wv.x;
    dp[(c + 1) * 2] = wv.y;
    dp[(c + 2) * 2] = wv.z;
    dp[(c + 3) * 2] = wv.w;
  }
  __syncthreads();

  const int wave = tid >> 5;
  const int lane = tid & 31;
  const int half = lane >> 4;   // lanes 16-31 hold K+2..K+3
  const int l16  = lane & 15;
  const int row0 = blockIdx.x * 64 + wave * 16;

  int arow = row0 + l16;
  if (arow >= M) arow = M - 1;
  const float* xrow = x + (size_t)arow * FIN + half * 2;

  v8f acc[8];
#pragma unroll
  for (int t = 0; t < 8; ++t) acc[t] = (v8f)0.f;

  for (int kb = 0; kb < FIN; kb += 4) {
    // A 16x4 f32 fragment: lanes 0-15 -> K = kb,kb+1 ; lanes 16-31 -> kb+2,kb+3
    v2f a = *(const v2f*)(xrow + kb);
    // B fragments: kb2 row = kb/2 + half, one b64 per tile
    const float* wb = sW + ((kb >> 1) + half) * 256 + l16 * 2;
#pragma unroll
    for (int t = 0; t < 8; ++t) {
      v2f b = *(const v2f*)(wb + t * 32);
      acc[t] = __builtin_amdgcn_wmma_f32_16x16x4_f32(
          false, a, false, b, (short)0, acc[t], false, false);
    }
  }

  // C/D: lanes 0-15 -> M=i, N=lane ; lanes 16-31 -> M=8+i, N=lane-16
  const int rbase = row0 + half * 8;
#pragma unroll
  for (int t = 0; t < 8; ++t) {
    float* op = h + (size_t)rbase * HC_ + t * 16 + l16;
#pragma unroll
    for (int i = 0; i < 8; ++i) {
      int r = rbase + i;
      if (r < M) op[(size_t)i * HC_] = acc[t][i];
    }
  }
}

// ---------------------------------------------------------------------------
// Kernel 2: per (bn, head): a_s = <h, att_src>, a_d = <h, att_dst> over C=32
// ---------------------------------------------------------------------------
__global__ void k_attn(const float* __restrict__ h, const float* __restrict__ att_src,
                       const float* __restrict__ att_dst, float* __restrict__ a_s,
                       float* __restrict__ a_d, int BNH) {
  int t = blockIdx.x * blockDim.x + threadIdx.x;
  if (t >= BNH) return;
  int head = t & (HEADS - 1);
  const float4* hp = (const float4*)(h + (size_t)t * CH);
  const float4* sp = (const float4*)(att_src + head * CH);
  const float4* dp = (const float4*)(att_dst + head * CH);
  float ssum = 0.f, dsum = 0.f;
#pragma unroll
  for (int i = 0; i < CH / 4; ++i) {
    float4 hv = hp[i], sv = sp[i], dv = dp[i];
    ssum += hv.x * sv.x + hv.y * sv.y + hv.z * sv.z + hv.w * sv.w;
    dsum += hv.x * dv.x + hv.y * dv.y + hv.z * dv.z + hv.w * dv.w;
  }
  a_s[t] = ssum;
  a_d[t] = dsum;
}

// ---------------------------------------------------------------------------
// CSR build over dst (edges are identical for all batches).
// ---------------------------------------------------------------------------
__global__ void k_zero(int* __restrict__ counts, int N) {
  int t = blockIdx.x * blockDim.x + threadIdx.x;
  if (t < N) counts[t] = 0;
}

__global__ void k_hist(const long long* __restrict__ ei, int* __restrict__ counts,
                       int E, int N) {
  int t = blockIdx.x * blockDim.x + threadIdx.x;
  if (t >= E + N) return;
  int dst = (t < E) ? (int)ei[(size_t)E + t] : (t - E);
  atomicAdd(&counts[dst], 1);
}

// Single-block scan: exclusive offsets over counts (N up to ~1e5 fine).
__global__ __launch_bounds__(1024) void k_scan(const int* __restrict__ counts,
                                               int* __restrict__ off,
                                               int* __restrict__ cursor, int N) {
  __shared__ int s[1024];
  __shared__ int carryS;
  int tid = threadIdx.x;
  if (tid == 0) carryS = 0;
  __syncthreads();
  for (int base = 0; base < N; base += 1024) {
    int i = base + tid;
    int v = (i < N) ? counts[i] : 0;
    s[tid] = v;
    __syncthreads();
    for (int o = 1; o < 1024; o <<= 1) {
      int t = (tid >= o) ? s[tid - o] : 0;
      __syncthreads();
      s[tid] += t;
      __syncthreads();
    }
    int incl = s[tid];
    int carry = carryS;
    int excl = carry + incl - v;
    if (i < N) { off[i] = excl; cursor[i] = excl; }
    __syncthreads();
    if (tid == 1023) carryS = carry + incl;
    __syncthreads();
  }
  if (tid == 0) off[N] = carryS;
}

__global__ void k_fill(const long long* __restrict__ ei, int* __restrict__ cursor,
                       int* __restrict__ csr, int E, int N) {
  int t = blockIdx.x * blockDim.x + threadIdx.x;
  if (t >= E + N) return;
  int src, dst;
  if (t < E) { src = (int)ei[t]; dst = (int)ei[(size_t)E + t]; }
  else       { src = dst = t - E; }
  int pos = atomicAdd(&cursor[dst], 1);
  csr[pos] = src;
}

// Canonicalize each dst segment (sort by src) -> deterministic summation order.
__global__ void k_sort(const int* __restrict__ off, int* __restrict__ csr, int N) {
  int d = blockIdx.x * blockDim.x + threadIdx.x;
  if (d >= N) return;
  int s = off[d], e = off[d + 1];
  for (int i = s + 1; i < e; ++i) {
    int key = csr[i];
    int j = i - 1;
    while (j >= s && csr[j] > key) { csr[j + 1] = csr[j]; --j; }
    csr[j + 1] = key;
  }
}

// ---------------------------------------------------------------------------
// Kernel 8: one wave per (dst, batch). Phase 1: segment max + denom per head
// via lane-strided loops + shfl_xor reduction. Phase 2: sequential edge loop,
// each lane owns 4 output channels; single float4 store. No float atomics.
// ---------------------------------------------------------------------------
__global__ __launch_bounds__(256) void k_gat(const int* __restrict__ off,
                                             const int* __restrict__ csr,
                                             const float* __restrict__ a_s,
                                             const float* __restrict__ a_d,
                                             const float* __restrict__ h,
                                             const float* __restrict__ bias,
                                             float* __restrict__ out, int N) {
  const int wave = threadIdx.x >> 5;
  const int lane = threadIdx.x & 31;
  const int dst  = blockIdx.x * 8 + wave;
  if (dst >= N) return;
  const int b = blockIdx.y;
  const size_t bnD = (size_t)b * N + dst;
  const int s0 = off[dst], s1 = off[dst + 1];

  float4 adv = *(const float4*)(a_d + bnD * HEADS);

  // --- phase 1a: per-head max over incoming edges ---
  float mx0 = -__builtin_inff(), mx1 = mx0, mx2 = mx0, mx3 = mx0;
  for (int i = s0 + lane; i < s1; i += 32) {
    int src = csr[i];
    float4 asv = *(const float4*)(a_s + ((size_t)b * N + src) * HEADS);
    mx0 = fmaxf(mx0, lrelu(asv.x + adv.x));
    mx1 = fmaxf(mx1, lrelu(asv.y + adv.y));
    mx2 = fmaxf(mx2, lrelu(asv.z + adv.z));
    mx3 = fmaxf(mx3, lrelu(asv.w + adv.w));
  }
#pragma unroll
  for (int o = 16; o >= 1; o >>= 1) {
    mx0 = fmaxf(mx0, __shfl_xor(mx0, o, 32));
    mx1 = fmaxf(mx1, __shfl_xor(mx1, o, 32));
    mx2 = fmaxf(mx2, __shfl_xor(mx2, o, 32));
    mx3 = fmaxf(mx3, __shfl_xor(mx3, o, 32));
  }
  // m = where(isfinite(m), m, 0)
  if (!(mx0 > -1e38f)) mx0 = 0.f;
  if (!(mx1 > -1e38f)) mx1 = 0.f;
  if (!(mx2 > -1e38f)) mx2 = 0.f;
  if (!(mx3 > -1e38f)) mx3 = 0.f;

  // --- phase 1b: per-head sum of exp(e - m) ---
  float sm0 = 0.f, sm1 = 0.f, sm2 = 0.f, sm3 = 0.f;
  for (int i = s0 + lane; i < s1; i += 32) {
    int src = csr[i];
    float4 asv = *(const float4*)(a_s + ((size_t)b * N + src) * HEADS);
    sm0 += __expf(lrelu(asv.x + adv.x) - mx0);
    sm1 += __expf(lrelu(asv.y + adv.y) - mx1);
    sm2 += __expf(lrelu(asv.z + adv.z) - mx2);
    sm3 += __expf(lrelu(asv.w + adv.w) - mx3);
  }
#pragma unroll
  for (int o = 16; o >= 1; o >>= 1) {
    sm0 += __shfl_xor(sm0, o, 32);
    sm1 += __shfl_xor(sm1, o, 32);
    sm2 += __shfl_xor(sm2, o, 32);
    sm3 += __shfl_xor(sm3, o, 32);
  }

  // --- phase 2: weighted gather; lane owns channels [lane*4, lane*4+4) ---
  const int head = lane >> 3;
  const float mh  = (head & 2) ? ((head & 1) ? mx3 : mx2) : ((head & 1) ? mx1 : mx0);
  const float dh  = ((head & 2) ? ((head & 1) ? sm3 : sm2) : ((head & 1) ? sm1 : sm0)) + 1e-16f;
  const float adh = (head & 2) ? ((head & 1) ? adv.w : adv.z) : ((head & 1) ? adv.y : adv.x);

  float ax = 0.f, ay = 0.f, az = 0.f, aw = 0.f;
  for (int i = s0; i < s1; ++i) {
    int src = csr[i];
    size_t bnS = (size_t)b * N + src;
    float e = lrelu(a_s[bnS * HEADS + head] + adh);
    float alpha = __expf(e - mh) / dh;
    float4 hv = *(const float4*)(h + bnS * HC_ + lane * 4);
    ax += alpha * hv.x;
    ay += alpha * hv.y;
    az += alpha * hv.z;
    aw += alpha * hv.w;
  }
  float4 bv = *(const float4*)(bias + lane * 4);
  float4 res; res.x = ax + bv.x; res.y = ay + bv.y; res.z = az + bv.z; res.w = aw + bv.w;
  *(float4*)(out + bnD * HC_ + lane * 4) = res;
}

extern "C" void kernel_launch(void* const* d_in, const int* in_sizes, int n_in,
                              void* d_out, int out_size, void* d_ws, size_t ws_size,
                              hipStream_t stream) {
  const float*     x       = (const float*)d_in[0];
  const long long* ei      = (const long long*)d_in[1];   // int64 per reference
  const float*     W       = (const float*)d_in[2];
  const float*     att_src = (const float*)d_in[3];
  const float*     att_dst = (const float*)d_in[4];
  const float*     bias    = (const float*)d_in[5];
  float*           out     = (float*)d_out;

  const int B = 4, N = 50000;
  const int E   = in_sizes[1] / 2;
  const int BN  = B * N;
  const int BNH = BN * HEADS;
  const int EN  = E + N;

  // Workspace: h (102.4MB) | a_s | a_d (3.2MB ea) | counts | off | cursor | csr (~7.2MB)
  float* h      = (float*)d_ws;
  float* a_s    = h + (size_t)BN * HC_;
  float* a_d    = a_s + BNH;
  int*   counts = (int*)(a_d + BNH);
  int*   off    = counts + N;
  int*   cursor = off + N + 1;
  int*   csr    = cursor + N;

  k_gemm<<<dim3((BN + 63) / 64), dim3(128), 0, stream>>>(x, W, h, BN);
  k_attn<<<dim3((BNH + 255) / 256), dim3(256), 0, stream>>>(h, att_src, att_dst, a_s, a_d, BNH);

  k_zero<<<dim3((N + 255) / 256), dim3(256), 0, stream>>>(counts, N);
  k_hist<<<dim3((EN + 255) / 256), dim3(256), 0, stream>>>(ei, counts, E, N);
  k_scan<<<dim3(1), dim3(1024), 0, stream>>>(counts, off, cursor, N);
  k_fill<<<dim3((EN + 255) / 256), dim3(256), 0, stream>>>(ei, cursor, csr, E, N);
  k_sort<<<dim3((N + 255) / 256), dim3(256), 0, stream>>>(off, csr, N);

  dim3 ggrid((N + 7) / 8, B);
  k_gat<<<ggrid, dim3(256), 0, stream>>>(off, csr, a_s, a_d, h, bias, out, N);
}